// TripletLoss_Balanced_2branch_10849087390448
// MI455X (gfx1250) — compile-verified
//
#include <hip/hip_runtime.h>
#include <hip/hip_bf16.h>

typedef __attribute__((ext_vector_type(2))) float v2f;
typedef __attribute__((ext_vector_type(8))) float v8f;

#define NN 128
#define DD 2048
#define HALF_NEG 56
#define MARGIN 0.3f
#define EPSV 1e-12f

// ---------------------------------------------------------------------------
// Kernel 1: row norms. blocks 0..127 -> xx (feat1), blocks 128..255 -> yy.
// ---------------------------------------------------------------------------
__global__ __launch_bounds__(256)
void norms_kernel(const float* __restrict__ f1, const float* __restrict__ f2,
                  float* __restrict__ xx, float* __restrict__ yy) {
    __shared__ float sred[256];
    const int row = blockIdx.x;
    const float* src = (row < NN) ? (f1 + (size_t)row * DD)
                                  : (f2 + (size_t)(row - NN) * DD);
    float s = 0.0f;
    for (int d = threadIdx.x; d < DD; d += 256) {
        float v = src[d];
        s += v * v;
    }
    sred[threadIdx.x] = s;
    __syncthreads();
    for (int off = 128; off > 0; off >>= 1) {
        if ((int)threadIdx.x < off) sred[threadIdx.x] += sred[threadIdx.x + off];
        __syncthreads();
    }
    if (threadIdx.x == 0) {
        if (row < NN) xx[row] = sred[0];
        else          yy[row - NN] = sred[0];
    }
}

// ---------------------------------------------------------------------------
// Kernel 2: 128x128 fp32 Gram via V_WMMA_F32_16X16X4_F32, fused distance
// epilogue: dist = sqrt(max(xx + yy - 2*dot, eps)). One 16x16 tile per wave,
// 64 waves total, 4 independent accumulators to pipeline the XDL.
// ---------------------------------------------------------------------------
__global__ __launch_bounds__(512)
void wmma_dist_kernel(const float* __restrict__ f1, const float* __restrict__ f2,
                      const float* __restrict__ xx, const float* __restrict__ yy,
                      float* __restrict__ dist) {
    const int wave = (int)((blockIdx.x * blockDim.x + threadIdx.x) >> 5); // 0..63
    const int lane = (int)(threadIdx.x & 31);
    const int I = wave >> 3;       // row tile
    const int J = wave & 7;        // col tile
    const int m = lane & 15;
    const int half = lane >> 4;

    // A fragment: lane m holds A[m][k + 2*half + {0,1}]   (16x4, f32)
    // B fragment: lane n holds B[k + 2*half + {0,1}][n] = feat2[n][k + 2*half + {0,1}]
    const float* __restrict__ arow = f1 + (size_t)(I * 16 + m) * DD + 2 * half;
    const float* __restrict__ brow = f2 + (size_t)(J * 16 + m) * DD + 2 * half;

    v8f acc0 = {}; v8f acc1 = {}; v8f acc2 = {}; v8f acc3 = {};
    for (int k = 0; k < DD; k += 16) {
        v2f a0 = *(const v2f*)(arow + k);
        v2f b0 = *(const v2f*)(brow + k);
        v2f a1 = *(const v2f*)(arow + k + 4);
        v2f b1 = *(const v2f*)(brow + k + 4);
        v2f a2 = *(const v2f*)(arow + k + 8);
        v2f b2 = *(const v2f*)(brow + k + 8);
        v2f a3 = *(const v2f*)(arow + k + 12);
        v2f b3 = *(const v2f*)(brow + k + 12);
        acc0 = __builtin_amdgcn_wmma_f32_16x16x4_f32(false, a0, false, b0, (short)0, acc0, false, false);
        acc1 = __builtin_amdgcn_wmma_f32_16x16x4_f32(false, a1, false, b1, (short)0, acc1, false, false);
        acc2 = __builtin_amdgcn_wmma_f32_16x16x4_f32(false, a2, false, b2, (short)0, acc2, false, false);
        acc3 = __builtin_amdgcn_wmma_f32_16x16x4_f32(false, a3, false, b3, (short)0, acc3, false, false);
    }
    v8f acc = (acc0 + acc1) + (acc2 + acc3);

    // C/D layout: VGPR r, lane l -> row = r + 8*(l>>4), col = l&15
    const int col = J * 16 + m;
    const float yyc = yy[col];
#pragma unroll
    for (int r = 0; r < 8; ++r) {
        const int row = I * 16 + r + 8 * half;
        float d2 = xx[row] + yyc - 2.0f * acc[r];
        d2 = fmaxf(d2, EPSV);
        dist[row * NN + col] = sqrtf(d2);
    }
}

// ---------------------------------------------------------------------------
// Kernel 3: hard example mining. One thread per anchor row; sequential scan
// reproduces stable-argsort (negatives in ascending column order) and
// argmin first-tie semantics of the reference.
// ---------------------------------------------------------------------------
__global__ __launch_bounds__(128)
void mining_kernel(const float* __restrict__ dist, const int* __restrict__ target,
                   float* __restrict__ dist_ap, int* __restrict__ sel1,
                   int* __restrict__ sel2) {
    const int i = (int)threadIdx.x;
    const int lab = target[i];
    float ap = -3.402823466e38f;
    float m1 = 3.402823466e38f, m2 = 3.402823466e38f;
    int s1 = 0, s2 = 0, negcnt = 0;
    for (int j = 0; j < NN; ++j) {
        const float dv = dist[i * NN + j];
        if (target[j] == lab) {
            ap = fmaxf(ap, dv);
        } else {
            if (negcnt < HALF_NEG) {
                if (dv < m1) { m1 = dv; s1 = j; }
            } else {
                if (dv < m2) { m2 = dv; s2 = j; }
            }
            ++negcnt;
        }
    }
    dist_ap[i] = ap;
    sel1[i] = s1;
    sel2[i] = s2;
}

// ---------------------------------------------------------------------------
// Kernel 4: dist_an per anchor. Anchor i pairs with feat_mean[map(i)] where
// map(i) = i (i<32), i+32 (32..63), i-32 (64..95), i (96..127), and
// feat_mean[k] = 0.5*(feat1[sel1[k]] + feat1[sel2[k]]).
// row_dist = sqrt(max(||a||^2 + ||b||^2 - 2 a.b, eps)).
// ---------------------------------------------------------------------------
__global__ __launch_bounds__(256)
void an_kernel(const float* __restrict__ f1, const int* __restrict__ sel1,
               const int* __restrict__ sel2, float* __restrict__ dist_an) {
    __shared__ float saa[256];
    __shared__ float sbb[256];
    __shared__ float sab[256];
    const int i = (int)blockIdx.x;
    const int k = (i < 32) ? i : (i < 64) ? i + 32 : (i < 96) ? i - 32 : i;
    const float* __restrict__ a = f1 + (size_t)i * DD;
    const float* __restrict__ u = f1 + (size_t)sel1[k] * DD;
    const float* __restrict__ v = f1 + (size_t)sel2[k] * DD;
    float aa = 0.0f, bb = 0.0f, ab = 0.0f;
    for (int d = threadIdx.x; d < DD; d += 256) {
        const float av = a[d];
        const float bv = 0.5f * (u[d] + v[d]);
        aa += av * av;
        bb += bv * bv;
        ab += av * bv;
    }
    saa[threadIdx.x] = aa;
    sbb[threadIdx.x] = bb;
    sab[threadIdx.x] = ab;
    __syncthreads();
    for (int off = 128; off > 0; off >>= 1) {
        if ((int)threadIdx.x < off) {
            saa[threadIdx.x] += saa[threadIdx.x + off];
            sbb[threadIdx.x] += sbb[threadIdx.x + off];
            sab[threadIdx.x] += sab[threadIdx.x + off];
        }
        __syncthreads();
    }
    if (threadIdx.x == 0) {
        float d2 = saa[0] + sbb[0] - 2.0f * sab[0];
        dist_an[i] = sqrtf(fmaxf(d2, EPSV));
    }
}

// ---------------------------------------------------------------------------
// Kernel 5: final loss reduction.
// ---------------------------------------------------------------------------
__global__ __launch_bounds__(128)
void loss_kernel(const float* __restrict__ dist_ap, const float* __restrict__ dist_an,
                 float* __restrict__ out) {
    __shared__ float ssum[128];
    __shared__ int scnt[128];
    const int i = (int)threadIdx.x;
    const float t = dist_ap[i] - dist_an[i] + MARGIN;
    ssum[i] = (t > 0.0f) ? t : 0.0f;
    scnt[i] = (t > 0.0f) ? 1 : 0;
    __syncthreads();
    for (int off = 64; off > 0; off >>= 1) {
        if (i < off) {
            ssum[i] += ssum[i + off];
            scnt[i] += scnt[i + off];
        }
        __syncthreads();
    }
    if (i == 0) {
        out[0] = ssum[0] / 128.0f;
        out[1] = (float)scnt[0];
    }
}

// ---------------------------------------------------------------------------
// Launch
// ---------------------------------------------------------------------------
extern "C" void kernel_launch(void* const* d_in, const int* in_sizes, int n_in,
                              void* d_out, int out_size, void* d_ws, size_t ws_size,
                              hipStream_t stream) {
    const float* f1 = (const float*)d_in[0];
    const float* f2 = (const float*)d_in[1];
    const int* target = (const int*)d_in[2];
    float* out = (float*)d_out;

    // workspace layout (floats)
    float* ws = (float*)d_ws;
    float* dist_mat = ws;              // 128*128
    float* xx       = ws + 16384;      // 128
    float* yy       = ws + 16512;      // 128
    float* dist_ap  = ws + 16640;      // 128
    float* dist_an  = ws + 16768;      // 128
    int*   sel1     = (int*)(ws + 16896); // 128
    int*   sel2     = (int*)(ws + 17024); // 128

    norms_kernel<<<256, 256, 0, stream>>>(f1, f2, xx, yy);
    wmma_dist_kernel<<<4, 512, 0, stream>>>(f1, f2, xx, yy, dist_mat);
    mining_kernel<<<1, 128, 0, stream>>>(dist_mat, target, dist_ap, sel1, sel2);
    an_kernel<<<128, 256, 0, stream>>>(f1, sel1, sel2, dist_an);
    loss_kernel<<<1, 128, 0, stream>>>(dist_ap, dist_an, out);
}